// Diffusion_Model_55482387530030
// MI455X (gfx1250) — compile-verified
//
#include <hip/hip_runtime.h>
#include <math.h>

#define NN 10000
#define EE 160000
#define BB 8
#define TT 12
#define HP 64
#define HV 32
#define NB (NN*BB)      /* 80000  */
#define EB (EE*BB)      /* 1280000 */

typedef float v2f __attribute__((ext_vector_type(2)));
typedef float v8f __attribute__((ext_vector_type(8)));

static __device__ __forceinline__ v2f ld2(const float* p) { return *(const v2f*)p; }

// monotonic float -> uint mapping for atomic max on signed floats
static __device__ __forceinline__ unsigned ordf(float f) {
    unsigned u = __float_as_uint(f);
    return (u >> 31) ? ~u : (u | 0x80000000u);
}
static __device__ __forceinline__ float deordf(unsigned x) {
    unsigned u = (x & 0x80000000u) ? (x & 0x7FFFFFFFu) : ~x;
    return __uint_as_float(u);
}

// ---------------------------------------------------------------- init
__global__ void k_init(float* pred, unsigned* amax, float* den) {
    int i = blockIdx.x * blockDim.x + threadIdx.x;
    if (i < NB) { pred[i] = 0.f; amax[i] = 0u; den[i] = 0.f; }
}

// -------------------------------------------- z = feature @ fc_w^T (WMMA f32)
// one wave = one 16-row tile of the [NB,12] A matrix; 4 N-tiles of 16 -> 64 cols
__global__ void k_node_embed(const float* __restrict__ feat,
                             const float* __restrict__ fcw,
                             float* __restrict__ z) {
    const int lane = threadIdx.x & 31;
    const int wave = threadIdx.x >> 5;
    const int tile = blockIdx.x * 8 + wave;          // NB/16 = 5000 tiles
    const int m = lane & 15, half = lane >> 4;
    const int row0 = tile * 16;

    v8f acc0 = {}, acc1 = {}, acc2 = {}, acc3 = {};
#pragma unroll
    for (int ks = 0; ks < 3; ks++) {
        const int k0 = ks * 4;
        v2f a = ld2(feat + (size_t)(row0 + m) * TT + k0 + 2 * half);
        v2f b0 = ld2(fcw + (size_t)(0 * 16 + m) * TT + k0 + 2 * half);
        v2f b1 = ld2(fcw + (size_t)(1 * 16 + m) * TT + k0 + 2 * half);
        v2f b2 = ld2(fcw + (size_t)(2 * 16 + m) * TT + k0 + 2 * half);
        v2f b3 = ld2(fcw + (size_t)(3 * 16 + m) * TT + k0 + 2 * half);
        acc0 = __builtin_amdgcn_wmma_f32_16x16x4_f32(false, a, false, b0, (short)0, acc0, false, false);
        acc1 = __builtin_amdgcn_wmma_f32_16x16x4_f32(false, a, false, b1, (short)0, acc1, false, false);
        acc2 = __builtin_amdgcn_wmma_f32_16x16x4_f32(false, a, false, b2, (short)0, acc2, false, false);
        acc3 = __builtin_amdgcn_wmma_f32_16x16x4_f32(false, a, false, b3, (short)0, acc3, false, false);
    }
#pragma unroll
    for (int r = 0; r < 8; r++) {
        size_t zr = (size_t)(row0 + r + 8 * half) * HP + m;
        z[zr + 0]  = acc0[r];
        z[zr + 16] = acc1[r];
        z[zr + 32] = acc2[r];
        z[zr + 48] = acc3[r];
    }
}

// layernorm(32)->relu->dot(32)->sigmoid head over a 16x32 WMMA result
// returns 8 row scalars for this lane-half's rows (replicated across half)
static __device__ __forceinline__ void mlp_head(
    const v8f& a0raw, const v8f& a1raw, int m,
    const float* __restrict__ bias, const float* __restrict__ g,
    const float* __restrict__ bt, const float* __restrict__ w12, float b12,
    float* out8) {
    float bi0 = bias[m], bi1 = bias[16 + m];
    float g0 = g[m], g1 = g[16 + m];
    float t0 = bt[m], t1 = bt[16 + m];
    float w0 = w12[m], w1 = w12[16 + m];
#pragma unroll
    for (int r = 0; r < 8; r++) {
        float x0 = a0raw[r] + bi0, x1 = a1raw[r] + bi1;
        float s = x0 + x1, q = x0 * x0 + x1 * x1;
#pragma unroll
        for (int msk = 1; msk < 16; msk <<= 1) {
            s += __shfl_xor(s, msk, 16);
            q += __shfl_xor(q, msk, 16);
        }
        float mean = s * (1.f / 32.f);
        float var = q * (1.f / 32.f) - mean * mean;
        float rs = rsqrtf(var + 1e-5f);
        float v0 = fmaxf((x0 - mean) * rs * g0 + t0, 0.f);
        float v1 = fmaxf((x1 - mean) * rs * g1 + t1, 0.f);
        float d = v0 * w0 + v1 * w1;
#pragma unroll
        for (int msk = 1; msk < 16; msk <<= 1) d += __shfl_xor(d, msk, 16);
        out8[r] = 1.f / (1.f + expf(-(d + b12)));
    }
}

// --------------------- fused velocity MLPs (WMMA) + T + diffusion series + S
__global__ void k_edge_vel(const float* __restrict__ up, const float* __restrict__ dn,
                           const float* __restrict__ l11w, const float* __restrict__ l11b,
                           const float* __restrict__ g11, const float* __restrict__ b11,
                           const float* __restrict__ l12w, const float* __restrict__ l12b,
                           const float* __restrict__ l21w, const float* __restrict__ l21b,
                           const float* __restrict__ g21, const float* __restrict__ b21,
                           const float* __restrict__ l22w, const float* __restrict__ l22b,
                           const float* __restrict__ l3w, const float* __restrict__ l3b,
                           const float* __restrict__ alpha, const float* __restrict__ dist,
                           const int* __restrict__ src, const float* __restrict__ feat,
                           float* __restrict__ Tbuf, float* __restrict__ Sbuf) {
    __shared__ float sh[8][2][16];
    const int lane = threadIdx.x & 31;
    const int wave = threadIdx.x >> 5;
    const int tile = blockIdx.x * 8 + wave;          // EB/16 = 80000 tiles
    const int m = lane & 15, half = lane >> 4;
    const int row0 = tile * 16;

    v8f au0 = {}, au1 = {}, ad0 = {}, ad1 = {};
#pragma unroll
    for (int ks = 0; ks < 3; ks++) {
        const int k0 = ks * 4;
        v2f aU = ld2(up + (size_t)(row0 + m) * TT + k0 + 2 * half);
        v2f aD = ld2(dn + (size_t)(row0 + m) * TT + k0 + 2 * half);
        v2f bu0 = ld2(l11w + (size_t)(m) * TT + k0 + 2 * half);
        v2f bu1 = ld2(l11w + (size_t)(16 + m) * TT + k0 + 2 * half);
        v2f bd0 = ld2(l21w + (size_t)(m) * TT + k0 + 2 * half);
        v2f bd1 = ld2(l21w + (size_t)(16 + m) * TT + k0 + 2 * half);
        au0 = __builtin_amdgcn_wmma_f32_16x16x4_f32(false, aU, false, bu0, (short)0, au0, false, false);
        au1 = __builtin_amdgcn_wmma_f32_16x16x4_f32(false, aU, false, bu1, (short)0, au1, false, false);
        ad0 = __builtin_amdgcn_wmma_f32_16x16x4_f32(false, aD, false, bd0, (short)0, ad0, false, false);
        ad1 = __builtin_amdgcn_wmma_f32_16x16x4_f32(false, aD, false, bd1, (short)0, ad1, false, false);
    }

    float su[8], sd[8];
    mlp_head(au0, au1, m, l11b, g11, b11, l12w, l12b[0], su);
    mlp_head(ad0, ad1, m, l21b, g21, b21, l22w, l22b[0], sd);

    if (m == 0) {   // lanes 0 and 16: publish this half's 8 row scalars
#pragma unroll
        for (int r = 0; r < 8; r++) {
            sh[wave][0][half * 8 + r] = su[r];
            sh[wave][1][half * 8 + r] = sd[r];
        }
    }
    __syncthreads();

    if (lane < 16) {
        int row = row0 + lane;
        int e = row >> 3, b = row & 7;
        float u_ = sh[wave][0][lane], d_ = sh[wave][1][lane];
        float al = alpha[e];
        float vv = l3w[0] * u_ + l3w[1] * d_ + l3w[2] * al + l3b[0];
        float sp = fmaxf(vv, 0.f) + log1pf(expf(-fabsf(vv)));   // stable softplus
        float v = fminf(sp, 3.f);
        float T = dist[e] / (v + 1e-5f);
        float Ti = fminf(fmaxf(rintf(T * 0.1f), 0.f), 11.f);    // RNE round, clip
        float alc = fminf(fmaxf(al, 0.f), 1.f);
        float F = 1.f / (1.f + alc * T);
        int nI = 12 - (int)Ti;
        float omf = 1.f - F;
        float tab[TT];
        float p = 1.f;
#pragma unroll
        for (int i = 0; i < TT; i++) { tab[i] = p; p *= omf; }
        const float* fb = feat + (size_t)(src[e] * BB + b) * TT;
        float S = 0.f;
#pragma unroll
        for (int k = 0; k < TT; k++) {
            int ei = nI - 1 - k;
            if (ei >= 0) S += F * tab[ei] * fb[k];
        }
        Tbuf[row] = T;
        Sbuf[row] = S;
    }
}

// ---------------------- attention logit: LN(129) . attn_w, wave per (e,b) row
__global__ void k_attn(const float* __restrict__ z, const float* __restrict__ Tbuf,
                       const int* __restrict__ src, const int* __restrict__ dst,
                       const float* __restrict__ g, const float* __restrict__ bt,
                       const float* __restrict__ aw,
                       unsigned* amax, float* __restrict__ abuf) {
    const int lane = threadIdx.x & 31;
    const int wave = threadIdx.x >> 5;
    const int row = blockIdx.x * 8 + wave;           // EB rows
    const int e = row >> 3, b = row & 7;
    const float* zs = z + (size_t)(src[e] * BB + b) * HP;
    const float* zd = z + (size_t)(dst[e] * BB + b) * HP;
    __builtin_prefetch(zs, 0, 0);
    __builtin_prefetch(zd, 0, 0);
    float T = Tbuf[row];
    float x[4];
#pragma unroll
    for (int i = 0; i < 4; i++) {
        int j = lane * 4 + i;
        x[i] = (j < 64) ? zs[j] : zd[j - 64];
    }
    float s = x[0] + x[1] + x[2] + x[3];
    float q = x[0] * x[0] + x[1] * x[1] + x[2] * x[2] + x[3] * x[3];
    if (lane == 0) { s += T; q += T * T; }
#pragma unroll
    for (int msk = 1; msk < 32; msk <<= 1) {
        s += __shfl_xor(s, msk, 32);
        q += __shfl_xor(q, msk, 32);
    }
    const float inv = 1.f / 129.f;
    float mean = s * inv;
    float var = q * inv - mean * mean;
    float rs = rsqrtf(var + 1e-5f);
    float d = 0.f;
#pragma unroll
    for (int i = 0; i < 4; i++) {
        int j = lane * 4 + i;
        d += aw[j] * ((x[i] - mean) * rs * g[j] + bt[j]);
    }
    if (lane == 0) d += aw[128] * ((T - mean) * rs * g[128] + bt[128]);
#pragma unroll
    for (int msk = 1; msk < 32; msk <<= 1) d += __shfl_xor(d, msk, 32);
    if (lane == 0) {
        float a = (d > 0.f) ? d : 0.01f * d;         // leaky_relu 0.01
        abuf[row] = a;
        atomicMax(&amax[src[e] * BB + b], ordf(a));
    }
}

// ------------------------------------ exp(a - max) and segment-sum denominator
__global__ void k_exp(const float* abuf, const int* __restrict__ src,
                      const unsigned* __restrict__ amax, float* exbuf, float* den) {
    int row = blockIdx.x * blockDim.x + threadIdx.x;
    if (row >= EB) return;
    int e = row >> 3, b = row & 7;
    float mx = deordf(amax[src[e] * BB + b]);
    float ex = expf(abuf[row] - mx);
    exbuf[row] = ex;
    atomicAdd(&den[src[e] * BB + b], ex);
}

// ------------------------------------------- normalize + scatter to pred[dst]
__global__ void k_scatter(const float* __restrict__ exbuf, const float* __restrict__ den,
                          const float* __restrict__ Sbuf, const int* __restrict__ src,
                          const int* __restrict__ dst, float* pred) {
    int row = blockIdx.x * blockDim.x + threadIdx.x;
    if (row >= EB) return;
    int e = row >> 3, b = row & 7;
    float w = exbuf[row] / den[src[e] * BB + b];
    atomicAdd(&pred[dst[e] * BB + b], w * Sbuf[row]);
}

extern "C" void kernel_launch(void* const* d_in, const int* in_sizes, int n_in,
                              void* d_out, int out_size, void* d_ws, size_t ws_size,
                              hipStream_t stream) {
    (void)in_sizes; (void)n_in; (void)out_size; (void)ws_size;
    const float* feature    = (const float*)d_in[0];
    const float* upstream   = (const float*)d_in[1];
    const float* downstream = (const float*)d_in[2];
    const float* distance   = (const float*)d_in[3];
    const int*   src        = (const int*)d_in[4];
    const int*   dst        = (const int*)d_in[5];
    const float* alpha      = (const float*)d_in[6];
    const float* fc_w       = (const float*)d_in[7];
    const float* ln2_g      = (const float*)d_in[8];
    const float* ln2_b      = (const float*)d_in[9];
    const float* attn_w     = (const float*)d_in[10];
    const float* l11_w      = (const float*)d_in[11];
    const float* l11_b      = (const float*)d_in[12];
    const float* ln11_g     = (const float*)d_in[13];
    const float* ln11_b     = (const float*)d_in[14];
    const float* l12_w      = (const float*)d_in[15];
    const float* l12_b      = (const float*)d_in[16];
    const float* l21_w      = (const float*)d_in[17];
    const float* l21_b      = (const float*)d_in[18];
    const float* ln21_g     = (const float*)d_in[19];
    const float* ln21_b     = (const float*)d_in[20];
    const float* l22_w      = (const float*)d_in[21];
    const float* l22_b      = (const float*)d_in[22];
    const float* l3_w       = (const float*)d_in[23];
    const float* l3_b       = (const float*)d_in[24];

    float* ws = (float*)d_ws;
    float*    z    = ws;                          // NB*HP  = 5,120,000 f
    float*    Tbuf = z + (size_t)NB * HP;         // EB
    float*    Sbuf = Tbuf + EB;                   // EB
    float*    abuf = Sbuf + EB;                   // EB (reused as exbuf)
    unsigned* amax = (unsigned*)(abuf + EB);      // NB
    float*    den  = (float*)(amax + NB);         // NB
    float*    pred = (float*)d_out;               // NB

    k_init<<<(NB + 255) / 256, 256, 0, stream>>>(pred, amax, den);
    k_node_embed<<<(NB / 16) / 8, 256, 0, stream>>>(feature, fc_w, z);
    k_edge_vel<<<(EB / 16) / 8, 256, 0, stream>>>(
        upstream, downstream,
        l11_w, l11_b, ln11_g, ln11_b, l12_w, l12_b,
        l21_w, l21_b, ln21_g, ln21_b, l22_w, l22_b,
        l3_w, l3_b, alpha, distance, src, feature, Tbuf, Sbuf);
    k_attn<<<EB / 8, 256, 0, stream>>>(z, Tbuf, src, dst, ln2_g, ln2_b, attn_w, amax, abuf);
    k_exp<<<(EB + 255) / 256, 256, 0, stream>>>(abuf, src, amax, abuf, den);
    k_scatter<<<(EB + 255) / 256, 256, 0, stream>>>(abuf, den, Sbuf, src, dst, pred);
}